// Conv1d_NN_49400713838645
// MI455X (gfx1250) — compile-verified
//
#include <hip/hip_runtime.h>
#include <hip/hip_bf16.h>

typedef float v2f __attribute__((ext_vector_type(2)));
typedef float v8f __attribute__((ext_vector_type(8)));

#define BATCH 4
#define N_TOK 8192
#define C_IN  16
#define C_OUT 16
#define KNN   9

// ---------------------------------------------------------------------------
// Kernel 0: column squared-norms  ns[b*N + n] = sum_c x[b,c,n]^2
// ---------------------------------------------------------------------------
__global__ void ns_kernel(const float* __restrict__ x, float* __restrict__ ns) {
    int i = blockIdx.x * blockDim.x + threadIdx.x;   // i = b*N + n
    if (i >= BATCH * N_TOK) return;
    int b = i >> 13;
    int n = i & (N_TOK - 1);
    const float* xb = x + (size_t)b * C_IN * N_TOK + n;
    float s = 0.0f;
#pragma unroll
    for (int c = 0; c < C_IN; ++c) {
        float v = xb[(size_t)c * N_TOK];
        s = fmaf(v, v, s);
    }
    ns[i] = s;
}

// ---------------------------------------------------------------------------
// Kernel 1: fused  Gram(WMMA fp32) -> running top-9 -> gather -> conv -> out
// block = 128 threads (4 waves); each wave owns 32 tokens of one batch.
// grid  = 256 blocks (4 batches x 64 row-blocks of 128 tokens).
// ---------------------------------------------------------------------------
__global__ __launch_bounds__(128) void knn_conv_kernel(
    const float* __restrict__ x, const float* __restrict__ W,
    const float* __restrict__ bias, const float* __restrict__ ns,
    float* __restrict__ out) {

    __shared__ float sW[C_IN * KNN * C_OUT];   // transposed: [(c*9+k)*16 + o]
    __shared__ float sScore[4][32 * 17];       // per-wave 32x16 tile, padded

    const int tid    = threadIdx.x;
    const int wave   = tid >> 5;
    const int lane   = tid & 31;
    const int lane16 = lane & 15;
    const int half   = lane >> 4;              // 0 or 1 (wave halves)

    // Stage W into LDS, transposed so output channel is innermost/contiguous.
    for (int idx = tid; idx < C_IN * KNN * C_OUT; idx += blockDim.x) {
        int o  = idx & 15;
        int ck = idx >> 4;                     // c*9 + k
        sW[idx] = W[o * (C_IN * KNN) + ck];
    }
    __syncthreads();

    const int b      = blockIdx.x >> 6;
    const int rowblk = blockIdx.x & 63;
    const int r0     = rowblk * 128 + wave * 32;   // first token of this wave
    const int n_tok  = r0 + lane;                  // token owned by this lane

    const float* xb  = x  + (size_t)b * C_IN * N_TOK;
    const float* nsb = ns + (size_t)b * N_TOK;

    // A tiles (rows fixed for the whole sweep).
    // f32 16x4 A layout: lanes 0-15 -> K=0,1 ; lanes 16-31 -> K=2,3; M = lane%16.
    v2f A0[4], A1[4];
#pragma unroll
    for (int q = 0; q < 4; ++q) {
        int c0 = 4 * q + 2 * half;
        A0[q].x = xb[(size_t)(c0    ) * N_TOK + r0      + lane16];
        A0[q].y = xb[(size_t)(c0 + 1) * N_TOK + r0      + lane16];
        A1[q].x = xb[(size_t)(c0    ) * N_TOK + r0 + 16 + lane16];
        A1[q].y = xb[(size_t)(c0 + 1) * N_TOK + r0 + 16 + lane16];
    }

    // Running top-9 (ascending scores; stable w.r.t. column index for ties).
    float bd[KNN];
    int   bi[KNN];
#pragma unroll
    for (int k = 0; k < KNN; ++k) { bd[k] = 3.4e38f; bi[k] = 0; }

    float* myScore = &sScore[wave][0];

#pragma unroll 1
    for (int m0 = 0; m0 < N_TOK; m0 += 16) {
        // Prefetch next column tile (first channel pair of this half-wave).
        if (m0 + 16 < N_TOK)
            __builtin_prefetch(&xb[(size_t)(2 * half) * N_TOK + m0 + 16 + lane16]);

        // B tile: columns m0..m0+15 of the Gram matrix (mirror of A layout).
        v2f Bt[4];
#pragma unroll
        for (int q = 0; q < 4; ++q) {
            int c0 = 4 * q + 2 * half;
            Bt[q].x = xb[(size_t)(c0    ) * N_TOK + m0 + lane16];
            Bt[q].y = xb[(size_t)(c0 + 1) * N_TOK + m0 + lane16];
        }

        // dot tiles via 8x V_WMMA_F32_16X16X4_F32 (chained C accumulate, K=16)
        v8f acc0 = {};   // rows r0 .. r0+15
        v8f acc1 = {};   // rows r0+16 .. r0+31
#pragma unroll
        for (int q = 0; q < 4; ++q) {
            acc0 = __builtin_amdgcn_wmma_f32_16x16x4_f32(
                false, A0[q], false, Bt[q], (short)0, acc0, false, false);
            acc1 = __builtin_amdgcn_wmma_f32_16x16x4_f32(
                false, A1[q], false, Bt[q], (short)0, acc1, false, false);
        }

        // score(m) = |x_m|^2 - 2*dot  (monotone in distance for fixed token)
        float nscol = nsb[m0 + lane16];
#pragma unroll
        for (int v = 0; v < 8; ++v) {
            int row0 = v + 8 * half;                 // C-tile row for this lane
            myScore[(row0     ) * 17 + lane16] = fmaf(-2.0f, acc0[v], nscol);
            myScore[(row0 + 16) * 17 + lane16] = fmaf(-2.0f, acc1[v], nscol);
        }

        // Intra-wave LDS store->load ordering (CDNA5 split DS counter).
        asm volatile("s_wait_dscnt 0" ::: "memory");

        // Bulk-load this lane's 16 candidates first so the DS loads clause
        // (pairable into ds_load_2addr_b32) under one dscnt wait, instead of
        // paying serialized LDS latency inside the compare chain.
        float cand[16];
#pragma unroll
        for (int j = 0; j < 16; ++j)
            cand[j] = myScore[lane * 17 + j];

        // Scan candidates from registers.
#pragma unroll
        for (int j = 0; j < 16; ++j) {
            float cd = cand[j];
            int   cm = m0 + j;
            if (cm == n_tok) cd = -3.4e38f;          // self is always nearest
            if (cd < bd[KNN - 1]) {
                bd[KNN - 1] = cd; bi[KNN - 1] = cm;
#pragma unroll
                for (int t = KNN - 1; t > 0; --t) {
                    if (bd[t] < bd[t - 1]) {
                        float td = bd[t]; bd[t] = bd[t - 1]; bd[t - 1] = td;
                        int   ti = bi[t]; bi[t] = bi[t - 1]; bi[t - 1] = ti;
                    }
                }
            }
        }
    }

    // Gather + conv: out[b,o,n] = bias[o] + sum_{k,c} x[b,c,idx[k]] * W[o,c,k]
    float acc[C_OUT];
#pragma unroll
    for (int o = 0; o < C_OUT; ++o) acc[o] = bias[o];
#pragma unroll 1
    for (int k = 0; k < KNN; ++k) {
        int m = bi[k];
#pragma unroll
        for (int c = 0; c < C_IN; ++c) {
            float f = xb[(size_t)c * N_TOK + m];
            const float* wrow = &sW[(c * KNN + k) * C_OUT];
#pragma unroll
            for (int o = 0; o < C_OUT; ++o) acc[o] = fmaf(f, wrow[o], acc[o]);
        }
    }
#pragma unroll
    for (int o = 0; o < C_OUT; ++o)
        out[((size_t)b * C_OUT + o) * N_TOK + n_tok] = acc[o];
}

// ---------------------------------------------------------------------------
extern "C" void kernel_launch(void* const* d_in, const int* in_sizes, int n_in,
                              void* d_out, int out_size, void* d_ws, size_t ws_size,
                              hipStream_t stream) {
    (void)in_sizes; (void)n_in; (void)out_size; (void)ws_size;
    const float* x    = (const float*)d_in[0];
    const float* W    = (const float*)d_in[1];
    const float* bias = (const float*)d_in[2];
    float*       out  = (float*)d_out;
    float*       ns   = (float*)d_ws;            // BATCH*N_TOK floats = 128 KB

    ns_kernel<<<(BATCH * N_TOK + 255) / 256, 256, 0, stream>>>(x, ns);
    knn_conv_kernel<<<BATCH * (N_TOK / 128), 128, 0, stream>>>(x, W, bias, ns, out);
}